// CNNMEAE_43396349558857
// MI455X (gfx1250) — compile-verified
//
#include <hip/hip_runtime.h>
#include <math.h>

// ---------------------------------------------------------------------------
// MemAE forward for MI455X (gfx1250, wave32).
//  - Memory-module GEMMs AND conv2/conv3/conv4 (implicit GEMM) run on matrix
//    cores via V_WMMA_F32_16X16X4_F32 (fp32 in / fp32 acc: no precision
//    downgrade vs the fp32 reference).
//  - Implicit-GEMM conv is templated on all layer dims so the im2col index
//    unpack uses constant-divisor strength reduction instead of the runtime
//    division sequences seen in round-2 disasm.
//  - 4 waves per block (one 16x16 C tile per wave) to fill the WGP's 4 SIMDs.
//  - softmax row staging uses CDNA5 GLOBAL_LOAD_ASYNC_TO_LDS + s_wait_asynccnt.
// Workspace requirement: ~291 MB (see offsets below).
// ---------------------------------------------------------------------------

typedef __attribute__((ext_vector_type(2))) float v2f;
typedef __attribute__((ext_vector_type(8))) float v8f;

#define NUM_MEM  2048
#define FEAT     984
#define THRESH   (1.0f / 2048.0f)
#define EPS_SHR  1e-15f
#define BN_EPS   1e-5f
#define SLOPE    0.01f

static inline int LB(long t) { return (int)((t + 255) / 256); }

__device__ __forceinline__ int clampi(int v, int lo, int hi)
{ return v < lo ? lo : (v > hi ? hi : v); }

// ---------------- direct conv (NCHW, OIHW) + bias (conv1 only) --------------
__global__ void conv2d_bias_k(const float* __restrict__ x, const float* __restrict__ w,
                              const float* __restrict__ bias, float* __restrict__ y,
                              int N, int IC, int IH, int IW,
                              int OC, int OH, int OW,
                              int KH, int KW, int SH, int SW, int PH, int PW)
{
    long idx = (long)blockIdx.x * blockDim.x + threadIdx.x;
    long total = (long)N * OC * OH * OW;
    if (idx >= total) return;
    int ow = (int)(idx % OW); long t = idx / OW;
    int oh = (int)(t % OH); t /= OH;
    int oc = (int)(t % OC);
    int n  = (int)(t / OC);
    float acc = bias[oc];
    int ih0 = oh * SH - PH;
    int iw0 = ow * SW - PW;
    for (int ic = 0; ic < IC; ++ic) {
        const float* xp = x + ((long)n * IC + ic) * IH * IW;
        const float* wp = w + ((long)oc * IC + ic) * KH * KW;
        for (int kh = 0; kh < KH; ++kh) {
            int ih = ih0 + kh;
            if (ih < 0 || ih >= IH) continue;
            const float* xr = xp + (long)ih * IW;
            for (int kw = 0; kw < KW; ++kw) {
                int iw = iw0 + kw;
                if (iw < 0 || iw >= IW) continue;
                acc = fmaf(xr[iw], wp[kh * KW + kw], acc);
            }
        }
    }
    y[idx] = acc;
}

// ---------------- implicit-GEMM conv on matrix cores ------------------------
// C[M,OC] = im2col(x)[M,K] * W^T[K,OC],  M = N*OH*OW, K = IC*KH*KW.
// All dims compile-time so index unpack / masks strength-reduce to constants.
// Requirements: M % 64 == 0 (4 waves/block), K % 4 == 0. OC tile masked.
// A gather is branchless (clamp + mask multiply) so EXEC is all-ones at WMMA.
template<int IC, int IH, int IW, int OC, int OH, int OW,
         int KH, int KW, int SH, int SW, int PH, int PW>
__global__ void wmma_conv_k(const float* __restrict__ x, const float* __restrict__ wgt,
                            const float* __restrict__ bias, float* __restrict__ y)
{
    constexpr int K   = IC * KH * KW;
    constexpr int KHW = KH * KW;
    static_assert(K % 4 == 0, "K must be divisible by 4");

    int wave = threadIdx.x >> 5;
    int lane = threadIdx.x & 31;
    int hl = lane >> 4;
    int l  = lane & 15;
    int mtile = blockIdx.x * 4 + wave;
    int mrow  = mtile * 16 + l;                  // A row this lane feeds
    int oc0   = blockIdx.y * 16;

    // decompose A row -> (n, oh, ow);  m = (n*OH + oh)*OW + ow
    int ow = mrow % OW; int t = mrow / OW;
    int oh = t % OH;    int n = t / OH;
    int ih0 = oh * SH - PH, iw0 = ow * SW - PW;
    const float* xn = x + (long)n * (IC * IH * IW);

    // B column masking for OC < 16 tiles (conv4): multiply-mask, no divergence
    int  col    = oc0 + l;
    float cmask = (col < OC) ? 1.f : 0.f;
    int  ccol   = (col < OC) ? col : (OC - 1);
    const float* wb = wgt + (long)ccol * K + 2 * hl;   // B[k][col] = wgt[col*K + k]

    v8f c = {};
    for (int k = 0; k < K; k += 4) {
        int kk = k + 2 * hl;
        // A element 0: unpack kk -> (ic, kh, kw); constant divisors
        int ic0 = kk / KHW;  int r0 = kk - ic0 * KHW;
        int kh0 = r0 / KW;   int kw0 = r0 - kh0 * KW;
        int iha = ih0 + kh0, iwa = iw0 + kw0;
        float ma = (iha >= 0 && iha < IH && iwa >= 0 && iwa < IW) ? 1.f : 0.f;
        int ihac = clampi(iha, 0, IH - 1), iwac = clampi(iwa, 0, IW - 1);
        // A element 1
        int kk1 = kk + 1;
        int ic1 = kk1 / KHW; int r1 = kk1 - ic1 * KHW;
        int kh1 = r1 / KW;   int kw1 = r1 - kh1 * KW;
        int ihb = ih0 + kh1, iwb = iw0 + kw1;
        float mb = (ihb >= 0 && ihb < IH && iwb >= 0 && iwb < IW) ? 1.f : 0.f;
        int ihbc = clampi(ihb, 0, IH - 1), iwbc = clampi(iwb, 0, IW - 1);

        v2f a, b;
        a.x = xn[(ic0 * IH + ihac) * IW + iwac] * ma;
        a.y = xn[(ic1 * IH + ihbc) * IW + iwbc] * mb;
        b.x = wb[k]     * cmask;
        b.y = wb[k + 1] * cmask;
        c = __builtin_amdgcn_wmma_f32_16x16x4_f32(false, a, false, b, (short)0, c, false, false);
    }

    if (col < OC) {
        float bv = bias[col];
#pragma unroll
        for (int r = 0; r < 8; ++r) {
            int row = mtile * 16 + r + 8 * hl;
            int oww = row % OW; int t2 = row / OW;
            int ohh = t2 % OH;  int nn = t2 / OH;
            y[(((long)nn * OC + col) * OH + ohh) * OW + oww] = c[r] + bv;
        }
    }
}

// ---------------- transposed conv (gather form), skip-add fused -------------
__global__ void convt_skip_bias_k(const float* __restrict__ x, const float* __restrict__ skip,
                                  const float* __restrict__ w, const float* __restrict__ bias,
                                  float* __restrict__ y,
                                  int N, int IC, int IH, int IW,
                                  int OC, int OH, int OW,
                                  int KH, int KW, int SH, int SW, int PH, int PW)
{
    long idx = (long)blockIdx.x * blockDim.x + threadIdx.x;
    long total = (long)N * OC * OH * OW;
    if (idx >= total) return;
    int ow = (int)(idx % OW); long t = idx / OW;
    int oh = (int)(t % OH); t /= OH;
    int oc = (int)(t % OC);
    int n  = (int)(t / OC);
    float acc = bias[oc];
    for (int ic = 0; ic < IC; ++ic) {
        const float* xp = x    + ((long)n * IC + ic) * IH * IW;
        const float* sp = skip + ((long)n * IC + ic) * IH * IW;
        const float* wp = w    + ((long)ic * OC + oc) * KH * KW;
        for (int kh = 0; kh < KH; ++kh) {
            int th_ = oh + PH - kh;
            if (th_ < 0 || (th_ % SH) != 0) continue;
            int ih = th_ / SH;
            if (ih >= IH) continue;
            for (int kw = 0; kw < KW; ++kw) {
                int tw = ow + PW - kw;
                if (tw < 0 || (tw % SW) != 0) continue;
                int iw = tw / SW;
                if (iw >= IW) continue;
                long o = (long)ih * IW + iw;
                acc = fmaf(xp[o] + sp[o], wp[kh * KW + kw], acc);
            }
        }
    }
    y[idx] = acc;
}

// ---------------- BN statistics: one block per channel ----------------------
__global__ void bn_stats_k(const float* __restrict__ x, float* __restrict__ stats,
                           int N, int C, int HW)
{
    int c = blockIdx.x;
    long per = (long)N * HW;
    float s = 0.f, s2 = 0.f;
    for (long i = threadIdx.x; i < per; i += blockDim.x) {
        long n = i / HW, r = i - n * HW;
        float v = x[(n * C + c) * (long)HW + r];
        s += v; s2 += v * v;
    }
    __shared__ float sh[256], sh2[256];
    int t = threadIdx.x;
    sh[t] = s; sh2[t] = s2;
    __syncthreads();
    for (int st = 128; st > 0; st >>= 1) {
        if (t < st) { sh[t] += sh[t + st]; sh2[t] += sh2[t + st]; }
        __syncthreads();
    }
    if (t == 0) {
        float inv = 1.f / (float)per;
        float m = sh[0] * inv;
        float v = sh2[0] * inv - m * m;
        stats[2 * c]     = m;
        stats[2 * c + 1] = v < 0.f ? 0.f : v;
    }
}

__global__ void bn_lrelu_k(float* __restrict__ y, const float* __restrict__ stats,
                           const float* __restrict__ g, const float* __restrict__ b,
                           int C, long HW, long total)
{
    long idx = (long)blockIdx.x * blockDim.x + threadIdx.x;
    if (idx >= total) return;
    int c = (int)((idx / HW) % C);
    float m = stats[2 * c], v = stats[2 * c + 1];
    float val = (y[idx] - m) * rsqrtf(v + BN_EPS) * g[c] + b[c];
    y[idx] = val >= 0.f ? val : SLOPE * val;
}

// ---------------- row L2 norms ---------------------------------------------
__global__ void row_l2norm_k(const float* __restrict__ x, float* __restrict__ out, int K)
{
    int r = blockIdx.x;
    const float* p = x + (long)r * K;
    float s = 0.f;
    for (int i = threadIdx.x; i < K; i += blockDim.x) { float v = p[i]; s += v * v; }
    __shared__ float sh[256];
    int t = threadIdx.x;
    sh[t] = s; __syncthreads();
    for (int st = 128; st > 0; st >>= 1) {
        if (t < st) sh[t] += sh[t + st];
        __syncthreads();
    }
    if (t == 0) out[r] = sqrtf(sh[0]);
}

// ---------------- WMMA fp32 GEMM: sim = (z . mem^T) / max(|z||m|,1e-8) -----
// 4 waves per block, one 16x16 tile per wave.
__global__ void wmma_cosine_sim_k(const float* __restrict__ z, const float* __restrict__ mem,
                                  const float* __restrict__ zn, const float* __restrict__ mn,
                                  float* __restrict__ sim)
{
    int wave = threadIdx.x >> 5;
    int lane = threadIdx.x & 31;
    int m0 = blockIdx.x * 16;
    int n0 = (blockIdx.y * 4 + wave) * 16;
    int hl = lane >> 4;
    int l  = lane & 15;
    const float* za = z   + (long)(m0 + l) * FEAT + 2 * hl;
    const float* mb = mem + (long)(n0 + l) * FEAT + 2 * hl;   // B[k][n] = mem[n][k]
    v8f c = {};
    for (int k = 0; k < FEAT; k += 4) {
        v2f a, b;
        a.x = za[k]; a.y = za[k + 1];
        b.x = mb[k]; b.y = mb[k + 1];
        c = __builtin_amdgcn_wmma_f32_16x16x4_f32(false, a, false, b, (short)0, c, false, false);
    }
    float mnl = mn[n0 + l];
#pragma unroll
    for (int r = 0; r < 8; ++r) {
        int row = m0 + r + 8 * hl;
        float den = fmaxf(zn[row] * mnl, 1e-8f);
        sim[(long)row * NUM_MEM + n0 + l] = c[r] / den;
    }
}

// ---------------- WMMA fp32 GEMM: z_hat = w @ mem (N=984, partial tile) ----
__global__ void wmma_zhat_k(const float* __restrict__ w, const float* __restrict__ mem,
                            float* __restrict__ zhat)
{
    int wave = threadIdx.x >> 5;
    int lane = threadIdx.x & 31;
    int ntile = blockIdx.y * 4 + wave;
    if (ntile >= 62) return;                      // whole-wave exit, EXEC stays full
    int m0 = blockIdx.x * 16;
    int n0 = ntile * 16;
    int hl = lane >> 4;
    int l  = lane & 15;
    int col = n0 + l;
    float bmask = (col < FEAT) ? 1.f : 0.f;
    int   ccol  = (col < FEAT) ? col : (FEAT - 1);
    const float* wa = w + (long)(m0 + l) * NUM_MEM + 2 * hl;
    v8f c = {};
    for (int k = 0; k < NUM_MEM; k += 4) {
        v2f a, b;
        a.x = wa[k]; a.y = wa[k + 1];
        int k0 = k + 2 * hl;
        b.x = mem[(long)k0 * FEAT + ccol] * bmask;
        b.y = mem[(long)(k0 + 1) * FEAT + ccol] * bmask;
        c = __builtin_amdgcn_wmma_f32_16x16x4_f32(false, a, false, b, (short)0, c, false, false);
    }
    if (col < FEAT) {
#pragma unroll
        for (int r = 0; r < 8; ++r)
            zhat[(long)(m0 + r + 8 * hl) * FEAT + col] = c[r];
    }
}

// ---------------- softmax -> hard-shrink -> L1 renorm, one block per row ---
// Row staged into LDS once via CDNA5 async global->LDS copies; 3 passes read LDS.
__global__ void softmax_shrink_k(const float* __restrict__ sim, float* __restrict__ w)
{
    int r = blockIdx.x;
    const float* s = sim + (long)r * NUM_MEM;
    float* wr = w + (long)r * NUM_MEM;
    __shared__ float row[NUM_MEM];
    __shared__ float sh[256];
    int t = threadIdx.x;

    // async stage: one B32 per element, tracked by ASYNCcnt
    {
        unsigned ldsbase = (unsigned)(uintptr_t)(void*)row;
        for (int i = t; i < NUM_MEM; i += 256) {
            unsigned lds_off = ldsbase + (unsigned)i * 4u;
            unsigned long long gaddr = (unsigned long long)(uintptr_t)(s + i);
            asm volatile("global_load_async_to_lds_b32 %0, %1, off"
                         :: "v"(lds_off), "v"(gaddr) : "memory");
        }
        asm volatile("s_wait_asynccnt 0" ::: "memory");
    }
    __syncthreads();

    float mx = -3.4e38f;
    for (int i = t; i < NUM_MEM; i += 256) mx = fmaxf(mx, row[i]);
    sh[t] = mx; __syncthreads();
    for (int st = 128; st > 0; st >>= 1) { if (t < st) sh[t] = fmaxf(sh[t], sh[t + st]); __syncthreads(); }
    mx = sh[0]; __syncthreads();

    float sum = 0.f;
    for (int i = t; i < NUM_MEM; i += 256) sum += expf(row[i] - mx);
    sh[t] = sum; __syncthreads();
    for (int st = 128; st > 0; st >>= 1) { if (t < st) sh[t] += sh[t + st]; __syncthreads(); }
    float inv = 1.f / sh[0]; __syncthreads();

    float asum = 0.f;
    for (int i = t; i < NUM_MEM; i += 256) {
        float ws = expf(row[i] - mx) * inv;
        float d  = ws - THRESH;
        float v  = fmaxf(d, 0.f) * ws / (fabsf(d) + EPS_SHR);
        wr[i] = v;
        asum += fabsf(v);
    }
    sh[t] = asum; __syncthreads();
    for (int st = 128; st > 0; st >>= 1) { if (t < st) sh[t] += sh[t + st]; __syncthreads(); }
    float ia = 1.f / sh[0]; __syncthreads();
    for (int i = t; i < NUM_MEM; i += 256) wr[i] *= ia;
}

// ---------------- top-2 + compact/spreading losses, one block per row ------
__global__ void mem_losses_k(const float* __restrict__ w, const float* __restrict__ z,
                             const float* __restrict__ mem,
                             float* __restrict__ compact, float* __restrict__ spread)
{
    int r = blockIdx.x;
    int t = threadIdx.x;
    const float* wr = w + (long)r * NUM_MEM;

    float v1 = -3.4e38f, v2 = -3.4e38f;
    int   i1 = 0x7fffffff, i2 = 0x7fffffff;
    for (int i = t; i < NUM_MEM; i += 256) {
        float v = wr[i];
        if (v > v1 || (v == v1 && i < i1)) { v2 = v1; i2 = i1; v1 = v; i1 = i; }
        else if (v > v2 || (v == v2 && i < i2)) { v2 = v; i2 = i; }
    }
    __shared__ float sv1[256], sv2[256], sv3[256];
    __shared__ int   si1[256], si2[256];
    sv1[t] = v1; sv2[t] = v2; si1[t] = i1; si2[t] = i2;
    __syncthreads();
    for (int st = 128; st > 0; st >>= 1) {
        if (t < st) {
            float a1 = sv1[t], a2 = sv2[t]; int b1 = si1[t], b2 = si2[t];
            float c1 = sv1[t + st], c2 = sv2[t + st]; int d1 = si1[t + st], d2 = si2[t + st];
            if (c1 > a1 || (c1 == a1 && d1 < b1)) { a2 = a1; b2 = b1; a1 = c1; b1 = d1; }
            else if (c1 > a2 || (c1 == a2 && d1 < b2)) { a2 = c1; b2 = d1; }
            if (c2 > a1 || (c2 == a1 && d2 < b1)) { a2 = a1; b2 = b1; a1 = c2; b1 = d2; }
            else if (c2 > a2 || (c2 == a2 && d2 < b2)) { a2 = c2; b2 = d2; }
            sv1[t] = a1; sv2[t] = a2; si1[t] = b1; si2[t] = b2;
        }
        __syncthreads();
    }
    int p = si1[0], q = si2[0];
    __syncthreads();

    const float* zr = z   + (long)r * FEAT;
    const float* mp = mem + (long)p * FEAT;
    const float* mq = mem + (long)q * FEAT;
    float sc = 0.f, sp = 0.f, sn = 0.f;
    for (int i = t; i < FEAT; i += 256) {
        float zd = zr[i] - mp[i];
        sc += zd * zd;
        float e1 = zd + 1e-6f;              sp += e1 * e1;
        float e2 = (zr[i] - mq[i]) + 1e-6f; sn += e2 * e2;
    }
    sv1[t] = sc; sv2[t] = sp; sv3[t] = sn;
    __syncthreads();
    for (int st = 128; st > 0; st >>= 1) {
        if (t < st) { sv1[t] += sv1[t + st]; sv2[t] += sv2[t + st]; sv3[t] += sv3[t + st]; }
        __syncthreads();
    }
    if (t == 0) {
        atomicAdd(compact, sv1[0]);
        float dp = sqrtf(sv2[0]), dn = sqrtf(sv3[0]);
        atomicAdd(spread, fmaxf(dp - dn + 10.f, 0.f));
    }
}

__global__ void sigmoid_k(float* __restrict__ y, long total)
{
    long idx = (long)blockIdx.x * blockDim.x + threadIdx.x;
    if (idx >= total) return;
    y[idx] = 1.f / (1.f + expf(-y[idx]));
}

// ---------------------------------------------------------------------------
extern "C" void kernel_launch(void* const* d_in, const int* in_sizes, int n_in,
                              void* d_out, int out_size, void* d_ws, size_t ws_size,
                              hipStream_t stream)
{
    (void)in_sizes; (void)n_in; (void)out_size; (void)ws_size;
    const float* x      = (const float*)d_in[0];
    const float* ec1_w  = (const float*)d_in[1];  const float* ec1_b = (const float*)d_in[2];
    const float* ec2_w  = (const float*)d_in[3];  const float* ec2_b = (const float*)d_in[4];
    const float* ec3_w  = (const float*)d_in[5];  const float* ec3_b = (const float*)d_in[6];
    const float* ec4_w  = (const float*)d_in[7];  const float* ec4_b = (const float*)d_in[8];
    const float* ebn1_g = (const float*)d_in[9];  const float* ebn1_b = (const float*)d_in[10];
    const float* ebn2_g = (const float*)d_in[11]; const float* ebn2_b = (const float*)d_in[12];
    const float* ebn3_g = (const float*)d_in[13]; const float* ebn3_b = (const float*)d_in[14];
    const float* ebn4_g = (const float*)d_in[15]; const float* ebn4_b = (const float*)d_in[16];
    const float* dc1_w  = (const float*)d_in[17]; const float* dc1_b = (const float*)d_in[18];
    const float* dc2_w  = (const float*)d_in[19]; const float* dc2_b = (const float*)d_in[20];
    const float* dc3_w  = (const float*)d_in[21]; const float* dc3_b = (const float*)d_in[22];
    const float* dc4_w  = (const float*)d_in[23]; const float* dc4_b = (const float*)d_in[24];
    const float* dbn1_g = (const float*)d_in[25]; const float* dbn1_b = (const float*)d_in[26];
    const float* dbn2_g = (const float*)d_in[27]; const float* dbn2_b = (const float*)d_in[28];
    const float* dbn3_g = (const float*)d_in[29]; const float* dbn3_b = (const float*)d_in[30];
    const float* memv   = (const float*)d_in[31];

    // workspace layout (floats)
    float* ws   = (float*)d_ws;
    float* e1   = ws;                      // 256*8*9*500    =  9,216,000
    float* e2   = e1 + 9216000L;           // 256*64*5*250   = 20,480,000
    float* e3   = e2 + 20480000L;          // 256*64*3*125   =  6,144,000
    float* e4   = e3 + 6144000L;           // 256*8*1*123    =    251,904 (== z)
    float* g1   = e4 + 251904L;            //  6,144,000
    float* g2   = g1 + 6144000L;           // 20,480,000
    float* g3   = g2 + 20480000L;          //  9,216,000
    float* simb = g3 + 9216000L;           //    524,288
    float* zhat = simb + 524288L;          //    251,904
    float* stats= zhat + 251904L;          //        128
    float* zn   = stats + 128L;            //        256
    float* mn   = zn + 256L;               //      2,048

    // output layout (floats): rec_x | w | z | compact | spreading
    float* outp  = (float*)d_out;
    float* rec   = outp;                   // 13,824,000
    float* wout  = outp + 13824000L;       //    524,288
    float* zout  = outp + 14348288L;       //    251,904
    float* closs = outp + 14600192L;       //          1
    float* sloss = closs + 1;              //          1

    dim3 b256(256), b128(128);

    // ---------------- encoder ----------------
    { long tot = 256L*8*9*500;   // conv1: K=27 (not %4), OC=8 -> direct VALU
      conv2d_bias_k<<<LB(tot), b256, 0, stream>>>(x, ec1_w, ec1_b, e1, 256,3,18,1000, 8,9,500, 3,3,2,2,1,1);
      bn_stats_k<<<8, b256, 0, stream>>>(e1, stats, 256, 8, 9*500);
      bn_lrelu_k<<<LB(tot), b256, 0, stream>>>(e1, stats, ebn1_g, ebn1_b, 8, 9L*500, tot); }
    { long tot = 256L*64*5*250;  // conv2 on matrix cores: M=320000 K=72 OC=64
      wmma_conv_k<8,9,500, 64,5,250, 3,3, 2,2, 1,1>
          <<<dim3(5000, 4), b128, 0, stream>>>(e1, ec2_w, ec2_b, e2);
      bn_stats_k<<<64, b256, 0, stream>>>(e2, stats, 256, 64, 5*250);
      bn_lrelu_k<<<LB(tot), b256, 0, stream>>>(e2, stats, ebn2_g, ebn2_b, 64, 5L*250, tot); }
    { long tot = 256L*64*3*125;  // conv3 on matrix cores: M=96000 K=576 OC=64
      wmma_conv_k<64,5,250, 64,3,125, 3,3, 2,2, 1,1>
          <<<dim3(1500, 4), b128, 0, stream>>>(e2, ec3_w, ec3_b, e3);
      bn_stats_k<<<64, b256, 0, stream>>>(e3, stats, 256, 64, 3*125);
      bn_lrelu_k<<<LB(tot), b256, 0, stream>>>(e3, stats, ebn3_g, ebn3_b, 64, 3L*125, tot); }
    { long tot = 256L*8*1*123;   // conv4 on matrix cores: M=31488 K=576 OC=8 (masked tile)
      wmma_conv_k<64,3,125, 8,1,123, 3,3, 1,1, 0,0>
          <<<dim3(492, 1), b128, 0, stream>>>(e3, ec4_w, ec4_b, e4);
      bn_stats_k<<<8, b256, 0, stream>>>(e4, stats, 256, 8, 1*123);
      bn_lrelu_k<<<LB(tot), b256, 0, stream>>>(e4, stats, ebn4_g, ebn4_b, 8, 123L, tot); }

    // ---------------- memory module (WMMA fp32 GEMMs) ----------------
    row_l2norm_k<<<256,  b256, 0, stream>>>(e4,   zn, FEAT);
    row_l2norm_k<<<2048, b256, 0, stream>>>(memv, mn, FEAT);
    wmma_cosine_sim_k<<<dim3(16, 32), b128, 0, stream>>>(e4, memv, zn, mn, simb);
    softmax_shrink_k<<<256, b256, 0, stream>>>(simb, wout);
    wmma_zhat_k<<<dim3(16, 16), b128, 0, stream>>>(wout, memv, zhat);
    hipMemsetAsync(closs, 0, 2 * sizeof(float), stream);
    mem_losses_k<<<256, b256, 0, stream>>>(wout, e4, memv, closs, sloss);
    hipMemcpyAsync(zout, e4, 251904L * sizeof(float), hipMemcpyDeviceToDevice, stream);

    // ---------------- decoder (skip-add fused into ConvT reads) ----------------
    { long tot = 256L*64*3*125;
      convt_skip_bias_k<<<LB(tot), b256, 0, stream>>>(zhat, e4, dc1_w, dc1_b, g1, 256,8,1,123, 64,3,125, 3,3,1,1,0,0);
      bn_stats_k<<<64, b256, 0, stream>>>(g1, stats, 256, 64, 3*125);
      bn_lrelu_k<<<LB(tot), b256, 0, stream>>>(g1, stats, dbn1_g, dbn1_b, 64, 3L*125, tot); }
    { long tot = 256L*64*5*250;
      convt_skip_bias_k<<<LB(tot), b256, 0, stream>>>(g1, e3, dc2_w, dc2_b, g2, 256,64,3,125, 64,5,250, 2,3,2,2,1,1);
      bn_stats_k<<<64, b256, 0, stream>>>(g2, stats, 256, 64, 5*250);
      bn_lrelu_k<<<LB(tot), b256, 0, stream>>>(g2, stats, dbn2_g, dbn2_b, 64, 5L*250, tot); }
    { long tot = 256L*8*9*500;
      convt_skip_bias_k<<<LB(tot), b256, 0, stream>>>(g2, e2, dc3_w, dc3_b, g3, 256,64,5,250, 8,9,500, 2,3,2,2,1,1);
      bn_stats_k<<<8, b256, 0, stream>>>(g3, stats, 256, 8, 9*500);
      bn_lrelu_k<<<LB(tot), b256, 0, stream>>>(g3, stats, dbn3_g, dbn3_b, 8, 9L*500, tot); }
    { long tot = 256L*3*18*1000;
      convt_skip_bias_k<<<LB(tot), b256, 0, stream>>>(g3, e1, dc4_w, dc4_b, rec, 256,8,9,500, 3,18,1000, 3,3,2,2,1,1);
      sigmoid_k<<<LB(tot), b256, 0, stream>>>(rec, tot); }
}